// BDC_83958020702529
// MI455X (gfx1250) — compile-verified
//
#include <hip/hip_runtime.h>
#include <math.h>

// ---------------------------------------------------------------------------
// Batched distance-covariance double-centering, MI455X (gfx1250, wave32).
// Gram via V_WMMA_F32_16X16X4_F32 (the only f32-exact matrix op on CDNA5).
// Each wave computes a 16x64 strip (4 tiles): A fragment loaded once per
// k-step, reused across 4 B fragments -> 1.25 float2 loads per WMMA instead
// of 2. x (16 MB) lives in L2 (192 MB), so upper-tri tiles are recomputed in
// the output pass instead of spilling 105 MB of dcov to HBM.
// ---------------------------------------------------------------------------

typedef __attribute__((ext_vector_type(2))) float v2f;
typedef __attribute__((ext_vector_type(8))) float v8f;

#define BB      64
#define DIMD    640
#define NK      100            // H*W
#define NT      40             // DIMD / 16
#define NSTRIP  10             // NT / 4  (4 j-tiles per wave)
#define EPSV    1e-5f
#define OUTROW  205120         // DIMD*(DIMD+1)/2

// workspace layout (floats)
#define WS_DIAG    0
#define WS_PART    (BB * DIMD)                     // 40960
#define WS_ROWSUM  (WS_PART + BB * DIMD * NSTRIP)  // 450560
#define WS_TOT     (WS_ROWSUM + BB * DIMD)         // 491520

// --- kernel A: diag[b,i] = sum_k x[b,i,k]^2 --------------------------------
__global__ void bdc_diag(const float* __restrict__ x, float* __restrict__ diag) {
    int idx = blockIdx.x * blockDim.x + threadIdx.x;   // b*DIMD + i
    if (idx >= BB * DIMD) return;
    const float* row = x + (size_t)idx * NK;
    float s = 0.f;
#pragma unroll 4
    for (int k = 0; k < NK; ++k) s = fmaf(row[k], row[k], s);
    diag[idx] = s;
}

// --- 16x64 Gram strip: A rows i0..i0+15 vs 4 consecutive 16-row j-tiles ----
// A layout (16x4 f32): lane = 16*half + m ; VGPR v holds K = kb + 2*half + v.
// B of x*x^T mirrors A with m -> n, so identical addressing with j-row base.
__device__ __forceinline__ void gram_strip(const float* __restrict__ xi,
                                           const float* __restrict__ xj,
                                           int lrow, int half, v8f c[4]) {
    const float* ai = xi + (size_t)lrow * NK + 2 * half;
    const float* bj = xj + (size_t)lrow * NK + 2 * half;
#pragma unroll
    for (int t = 0; t < 4; ++t) c[t] = (v8f){};
#pragma unroll
    for (int kb = 0; kb < NK; kb += 4) {
        v2f a = *(const v2f*)(ai + kb);
#pragma unroll
        for (int t = 0; t < 4; ++t) {
            v2f b = *(const v2f*)(bj + (size_t)t * 16 * NK + kb);
            // (neg_a, A, neg_b, B, c_mod, C, reuse_a, reuse_b)
            c[t] = __builtin_amdgcn_wmma_f32_16x16x4_f32(false, a, false, b,
                                                         (short)0, c[t], false, false);
        }
    }
}

// --- kernel B: all 40x10 strips -> per-(row, strip) dcov partial sums ------
__global__ void bdc_partial(const float* __restrict__ x,
                            const float* __restrict__ temp,
                            const float* __restrict__ diag,
                            float* __restrict__ partial) {
    int wave = threadIdx.x >> 5;
    int lane = threadIdx.x & 31;
    int w    = blockIdx.x * 8 + wave;      // 0..399
    int b    = blockIdx.y;
    int ti = w / NSTRIP, strip = w % NSTRIP;
    int i0 = ti * 16,    j0 = strip * 64;
    int lrow = lane & 15, half = lane >> 4;

    const float* xb = x + (size_t)b * DIMD * NK;
    v8f c[4];
    gram_strip(xb + (size_t)i0 * NK, xb + (size_t)j0 * NK, lrow, half, c);

    float scale = expf(temp[0]);
    const float* db = diag + b * DIMD;
    float dj[4];
#pragma unroll
    for (int t = 0; t < 4; ++t) dj[t] = db[j0 + 16 * t + lrow];

#pragma unroll
    for (int r = 0; r < 8; ++r) {
        int i = i0 + r + 8 * half;         // C/D layout: VGPR r, halves +0/+8
        float di = db[i];
        float acc = 0.f;
#pragma unroll
        for (int t = 0; t < 4; ++t) {      // sum the 4 tiles in-register first
            float v = di + dj[t] - 2.0f * c[t][r];
            acc += sqrtf(scale * fmaxf(v, 0.0f) + EPSV);
        }
        // reduce the 16 column-lanes of this half (xor<=8 stays in half)
        acc += __shfl_xor(acc, 1);
        acc += __shfl_xor(acc, 2);
        acc += __shfl_xor(acc, 4);
        acc += __shfl_xor(acc, 8);
        if (lrow == 0)                     // unique writer -> deterministic
            partial[((size_t)b * DIMD + i) * NSTRIP + strip] = acc;
    }
}

// --- kernel C: rowsum[b,i] = sum_strip partial; tot[b] = sum_i rowsum ------
__global__ void bdc_reduce(const float* __restrict__ partial,
                           float* __restrict__ rowsum,
                           float* __restrict__ tot) {
    __shared__ float sdata[DIMD];
    int b = blockIdx.x;
    int d = threadIdx.x;                   // 640 threads
    const float* p = partial + ((size_t)b * DIMD + d) * NSTRIP;
    float s = 0.f;
#pragma unroll
    for (int t = 0; t < NSTRIP; ++t) s += p[t];
    rowsum[b * DIMD + d] = s;
    sdata[d] = s;
    __syncthreads();
    for (int stride = 512; stride > 0; stride >>= 1) {
        if (d < stride && d + stride < DIMD) sdata[d] += sdata[d + stride];
        __syncthreads();
    }
    if (d == 0) tot[b] = sdata[0];
}

// --- kernel D: recompute upper-tri strips, center, scatter to packed triu --
__global__ void bdc_out(const float* __restrict__ x,
                        const float* __restrict__ temp,
                        const float* __restrict__ diag,
                        const float* __restrict__ rowsum,
                        const float* __restrict__ tot,
                        float* __restrict__ out) {
    int wave = threadIdx.x >> 5;
    int lane = threadIdx.x & 31;
    int w    = blockIdx.x * 8 + wave;
    int b    = blockIdx.y;
    int ti = w / NSTRIP, strip = w % NSTRIP;
    int j0 = strip * 64;
    if (ti * 16 > j0 + 63) return;         // whole strip below diagonal (wave-uniform)
    int i0 = ti * 16;
    int lrow = lane & 15, half = lane >> 4;

    const float* xb = x + (size_t)b * DIMD * NK;
    v8f c[4];
    gram_strip(xb + (size_t)i0 * NK, xb + (size_t)j0 * NK, lrow, half, c);

    float scale = expf(temp[0]);
    const float inv_d = 1.0f / (float)DIMD;
    const float* db = diag   + b * DIMD;
    const float* rs = rowsum + b * DIMD;
    float tm = tot[b] * inv_d * inv_d;
    float* ob = out + (size_t)b * OUTROW;

    float dj[4], mj[4];
#pragma unroll
    for (int t = 0; t < 4; ++t) {
        int j = j0 + 16 * t + lrow;
        dj[t] = db[j];
        mj[t] = rs[j] * inv_d;
    }
#pragma unroll
    for (int r = 0; r < 8; ++r) {
        int i = i0 + r + 8 * half;
        float di = db[i];
        float mi = rs[i] * inv_d;
        int base = i * DIMD - (i * (i - 1)) / 2 - i;   // triu row start - i
#pragma unroll
        for (int t = 0; t < 4; ++t) {
            int j = j0 + 16 * t + lrow;
            float v = di + dj[t] - 2.0f * c[t][r];
            v = sqrtf(scale * fmaxf(v, 0.0f) + EPSV);
            v = v - mi - mj[t] + tm;       // double-centering (symmetric dcov)
            if (j >= i) ob[base + j] = v;
        }
    }
}

extern "C" void kernel_launch(void* const* d_in, const int* in_sizes, int n_in,
                              void* d_out, int out_size, void* d_ws, size_t ws_size,
                              hipStream_t stream) {
    const float* x    = (const float*)d_in[0];
    const float* temp = (const float*)d_in[1];
    float* out = (float*)d_out;
    float* ws  = (float*)d_ws;

    float* diag    = ws + WS_DIAG;
    float* partial = ws + WS_PART;
    float* rowsum  = ws + WS_ROWSUM;
    float* tot     = ws + WS_TOT;

    // 400 waves per batch, 8 waves (256 threads) per block -> 50 blocks/batch
    bdc_diag   <<<dim3((BB * DIMD + 255) / 256), 256, 0, stream>>>(x, diag);
    bdc_partial<<<dim3(50, BB), 256, 0, stream>>>(x, temp, diag, partial);
    bdc_reduce <<<dim3(BB), DIMD, 0, stream>>>(partial, rowsum, tot);
    bdc_out    <<<dim3(50, BB), 256, 0, stream>>>(x, temp, diag, rowsum, tot, out);
}